// GCNLayer_11708080849500
// MI455X (gfx1250) — compile-verified
//
#include <hip/hip_runtime.h>

typedef __attribute__((ext_vector_type(2))) float v2f;
typedef __attribute__((ext_vector_type(8))) float v8f;

#define D 128   // D_IN == D_OUT == 128

// ---------------------------------------------------------------- zero fill
__global__ void __launch_bounds__(256) gcn_zero_kernel(float* __restrict__ p, long n) {
    long i = (long)blockIdx.x * blockDim.x + threadIdx.x;
    if (i < n) p[i] = 0.0f;
}

// ---------------------------------------------------------------- degrees
__global__ void __launch_bounds__(256) gcn_degree_kernel(const int* __restrict__ snd,
                                                         const int* __restrict__ rcv,
                                                         float* __restrict__ deg_s,
                                                         float* __restrict__ deg_r,
                                                         int n_edges) {
    int i = blockIdx.x * blockDim.x + threadIdx.x;
    if (i < n_edges) {
        unsafeAtomicAdd(&deg_s[snd[i]], 1.0f);   // hardware global_atomic_add_f32
        unsafeAtomicAdd(&deg_r[rcv[i]], 1.0f);
    }
}

// ---------------------------------------------------------------- fp32 WMMA GEMM
// h[m, n] = (sum_k x[m,k] * W[k,n] + bias[n]) * rsqrt(max(deg_s[m], 1))
// Block: 256 threads = 8 waves. Block covers 16 node rows; wave w covers cols
// [16w, 16w+16). Wave accumulates a 16x16 f32 tile with V_WMMA_F32_16X16X4_F32.
__global__ void __launch_bounds__(256) gcn_gemm_scale_kernel(const float* __restrict__ x,
                                                             const float* __restrict__ W,
                                                             const float* __restrict__ bias,
                                                             const float* __restrict__ deg_s,
                                                             float* __restrict__ h,
                                                             int n_nodes) {
    const int lane = threadIdx.x & 31;
    const int wave = threadIdx.x >> 5;      // 0..7  -> N tile
    const int l16  = lane & 15;
    const int half = lane >> 4;             // 0: K k..k+1 | 1: K k+2..k+3
    const int m0   = blockIdx.x * 16;
    const int n0   = wave * 16;

    // A-matrix row for this lane (16x4 f32 layout, ISA 7.12.2):
    //   lanes 0-15 hold M=0..15 with V0=K0,V1=K1; lanes 16-31 hold V0=K2,V1=K3
    const float* __restrict__ arow = x + (size_t)(m0 + l16) * D;

    v8f c = {};
    #pragma unroll 8
    for (int k = 0; k < D; k += 4) {
        const int ka = k + 2 * half;
        v2f a, b;
        a[0] = arow[ka];
        a[1] = arow[ka + 1];
        // B-matrix 4x16: row K striped across lanes within a VGPR (mirror of A)
        b[0] = W[(size_t)ka * D + n0 + l16];
        b[1] = W[(size_t)(ka + 1) * D + n0 + l16];
        c = __builtin_amdgcn_wmma_f32_16x16x4_f32(
                /*neg_a=*/false, a, /*neg_b=*/false, b,
                /*c_mod=*/(short)0, c, /*reuse_a=*/false, /*reuse_b=*/false);
    }

    // Epilogue: C/D layout -> VGPR r holds row (r + 8*half), col l16
    const float bv = bias[n0 + l16];
    #pragma unroll
    for (int r = 0; r < 8; ++r) {
        const int m = m0 + r + 8 * half;
        const float s = rsqrtf(fmaxf(deg_s[m], 1.0f));
        h[(size_t)m * D + n0 + l16] = (c[r] + bv) * s;
    }
}

// ---------------------------------------------------------------- edge scatter
// 32 lanes per edge, float4 per lane (128 dims). h and agg are L2-resident
// (51 MB each, 192 MB L2), so random gather + f32 atomics stay on-chip.
__global__ void __launch_bounds__(256) gcn_scatter_kernel(const float* __restrict__ h,
                                                          const int* __restrict__ snd,
                                                          const int* __restrict__ rcv,
                                                          float* __restrict__ agg,
                                                          int n_edges) {
    const int e = blockIdx.x * 8 + (threadIdx.x >> 5);
    if (e >= n_edges) return;
    const int lane = threadIdx.x & 31;
    const int s = snd[e];
    const int r = rcv[e];
    const float4 v = ((const float4*)(h + (size_t)s * D))[lane];
    float* dst = agg + (size_t)r * D + lane * 4;
    unsafeAtomicAdd(dst + 0, v.x);
    unsafeAtomicAdd(dst + 1, v.y);
    unsafeAtomicAdd(dst + 2, v.z);
    unsafeAtomicAdd(dst + 3, v.w);
}

// ---------------------------------------------------------------- post scale
__global__ void __launch_bounds__(128) gcn_out_scale_kernel(float* __restrict__ out,
                                                            const float* __restrict__ deg_r) {
    const int m = blockIdx.x;
    const float s = rsqrtf(fmaxf(deg_r[m], 1.0f));
    out[(size_t)m * D + threadIdx.x] *= s;
}

// ---------------------------------------------------------------- launcher
extern "C" void kernel_launch(void* const* d_in, const int* in_sizes, int n_in,
                              void* d_out, int out_size, void* d_ws, size_t ws_size,
                              hipStream_t stream) {
    const float* x    = (const float*)d_in[0];
    const int*   snd  = (const int*)d_in[1];
    const int*   rcv  = (const int*)d_in[2];
    // d_in[3] = n_node scalar (unused; derive from sizes)
    const float* W    = (const float*)d_in[4];
    const float* bias = (const float*)d_in[5];

    const int n_nodes = in_sizes[0] / D;   // 100000
    const int n_edges = in_sizes[1];       // 1600000

    // Workspace layout: [deg_s | deg_r | h]
    float* deg_s = (float*)d_ws;
    float* deg_r = deg_s + n_nodes;
    float* h     = deg_r + n_nodes;
    float* out   = (float*)d_out;

    const long n_deg = 2L * n_nodes;
    gcn_zero_kernel<<<(int)((n_deg + 255) / 256), 256, 0, stream>>>(deg_s, n_deg);

    const long n_out = (long)n_nodes * D;
    gcn_zero_kernel<<<(int)((n_out + 255) / 256), 256, 0, stream>>>(out, n_out);

    gcn_degree_kernel<<<(n_edges + 255) / 256, 256, 0, stream>>>(snd, rcv, deg_s, deg_r, n_edges);

    gcn_gemm_scale_kernel<<<n_nodes / 16, 256, 0, stream>>>(x, W, bias, deg_s, h, n_nodes);

    gcn_scatter_kernel<<<(n_edges + 7) / 8, 256, 0, stream>>>(h, snd, rcv, out, n_edges);

    gcn_out_scale_kernel<<<n_nodes, 128, 0, stream>>>(out, deg_r);
}